// EnhancedHybridQNN_11269994185247
// MI455X (gfx1250) — compile-verified
//
#include <hip/hip_runtime.h>
#include <hip/hip_bf16.h>
#include <math.h>

typedef float v2f __attribute__((ext_vector_type(2)));
typedef float v8f __attribute__((ext_vector_type(8)));

#define BATCH 8192
#define QDIM  256
#define NQ    8
#define NL    6

// workspace layout (float offsets). Total ~459k floats (~1.8 MB).
#define WS_U    0          // 384   fused per-(layer,qubit) 2x2 complex unitaries
#define WS_CF   384        // 32768 feature-processor output (B,4)
#define WS_QZ   33152      // 32768 quantum Z-expectations (B,4)
#define WS_A1   65920      // 262144 pre-BN1 activations (B,32)
#define WS_ST1  328064     // 64: sum[32], sumsq[32]
#define WS_A2   328128     // 131072 pre-BN2 activations (B,16)
#define WS_ST2  459200     // 32: sum[16], sumsq[16]

__device__ __forceinline__ float2 cmul(float2 a, float2 b){
  return make_float2(a.x*b.x - a.y*b.y, a.x*b.y + a.y*b.x);
}
__device__ __forceinline__ float2 cadd(float2 a, float2 b){
  return make_float2(a.x+b.x, a.y+b.y);
}

// ---------------- K1: fuse RZ*RY*RX per (layer,qubit) ----------------
__global__ void k_fuse_rot(const float* __restrict__ qw, float* __restrict__ ws){
  int t = threadIdx.x;
  if (t >= NL*NQ) return;
  int l = t / NQ, q = t % NQ;
  float tx = 0.5f*qw[l*3*NQ + 3*q + 0];
  float ty = 0.5f*qw[l*3*NQ + 3*q + 1];
  float tz = 0.5f*qw[l*3*NQ + 3*q + 2];
  float cx=cosf(tx), sx=sinf(tx);
  float cy=cosf(ty), sy=sinf(ty);
  float cz=cosf(tz), sz=sinf(tz);
  // A = RY*RX with RX=[[c,-is],[-is,c]], RY=[[c,-s],[s,c]]
  float2 a00 = make_float2( cy*cx,  sy*sx);
  float2 a01 = make_float2(-sy*cx, -cy*sx);
  float2 a10 = make_float2( sy*cx, -cy*sx);
  float2 a11 = make_float2( cy*cx, -sy*sx);
  // U = RZ*A, RZ = diag(cz - i sz, cz + i sz)
  float2 e0 = make_float2(cz, -sz), e1 = make_float2(cz, sz);
  float2* U = (float2*)(ws + WS_U) + t*4;
  U[0] = cmul(e0, a00); U[1] = cmul(e0, a01);
  U[2] = cmul(e1, a10); U[3] = cmul(e1, a11);
}

// ---------------- K2: feature processor via WMMA f32 16x16x4 ----------------
// cf = tanh(relu(x @ fW1^T + fb1) @ fW2^T + fb2)
// one wave computes a 16-row strip; 4 waves/block -> 64 rows/block.
__global__ void __launch_bounds__(128) k_feature(const float* __restrict__ x,
    const float* __restrict__ fW1, const float* __restrict__ fb1,
    const float* __restrict__ fW2, const float* __restrict__ fb2,
    float* __restrict__ ws){
  __shared__ float lh[4][16][68];           // hidden tile per wave, padded
  const int lane = threadIdx.x & 31;
  const int wave = threadIdx.x >> 5;
  const int row0 = blockIdx.x*64 + wave*16;
  const int M    = lane & 15;
  const int half = lane >> 4;
  const int kb   = half*2;                  // K sub-offset within 4-wide K-block

  const v8f vz = {0.f,0.f,0.f,0.f,0.f,0.f,0.f,0.f};
  v8f acc[4] = {vz, vz, vz, vz};

  const float* xr = x + (size_t)(row0 + M)*QDIM + kb;
  for (int k=0; k<QDIM; k+=4){
    v2f a = *(const v2f*)(xr + k);          // A[M][k+kb .. k+kb+1]
    #pragma unroll
    for (int t=0; t<4; ++t){
      // B[K][N] = fW1^T[k+K][t*16+N] = fW1[t*16+N][k+K]; lane holds N=M, K=kb,kb+1
      v2f b = *(const v2f*)(fW1 + (size_t)(t*16+M)*QDIM + k + kb);
      acc[t] = __builtin_amdgcn_wmma_f32_16x16x4_f32(
          false, a, false, b, (short)0, acc[t], false, false);
    }
  }
  // bias + relu, stage to LDS in row-major for A-fragment re-read
  #pragma unroll
  for (int t=0; t<4; ++t){
    float bias = fb1[t*16 + M];
    #pragma unroll
    for (int r=0; r<8; ++r){
      float v = acc[t][r] + bias;           // row = r + half*8, col = t*16+M
      lh[wave][r + half*8][t*16 + M] = v > 0.f ? v : 0.f;
    }
  }
  __syncthreads();

  // GEMM2: (16x64) @ fW2^T (64x4), N padded to 16 with zeros
  v8f acc2 = vz;
  for (int k=0; k<64; k+=4){
    v2f a; a.x = lh[wave][M][k+kb]; a.y = lh[wave][M][k+kb+1];
    v2f b;
    b.x = (M < 4) ? fW2[M*64 + k + kb]     : 0.f;
    b.y = (M < 4) ? fW2[M*64 + k + kb + 1] : 0.f;
    acc2 = __builtin_amdgcn_wmma_f32_16x16x4_f32(
        false, a, false, b, (short)0, acc2, false, false);
  }
  if (M < 4){
    float bias = fb2[M];
    float* cf = ws + WS_CF;
    #pragma unroll
    for (int r=0; r<8; ++r){
      int row = row0 + r + half*8;
      cf[(size_t)row*4 + M] = tanhf(acc2[r] + bias);
    }
  }
}

// ---------------- K3: quantum statevector simulation ----------------
// CNOT chain composed into one GF(2)-linear index gather per layer.
__device__ __forceinline__ unsigned cnot_src(unsigned j){
  // application order: (0,1)(1,2)(2,3)(3,4)(4,5)(5,6)(6,7)(7,0)(0,2)(2,4)(4,6)
  // source index applies the (self-inverse) gates in REVERSE order. bit(w)=7-w.
  const int C[11][2] = {{4,6},{2,4},{0,2},{7,0},{6,7},{5,6},{4,5},{3,4},{2,3},{1,2},{0,1}};
  #pragma unroll
  for (int i=0; i<11; ++i){
    unsigned bc = 7u - (unsigned)C[i][0], bt = 7u - (unsigned)C[i][1];
    j ^= ((j >> bc) & 1u) << bt;
  }
  return j;
}

__global__ void __launch_bounds__(256) k_quantum(const float* __restrict__ x,
                                                 float* __restrict__ ws){
  __shared__ float2 st[8][QDIM];            // 16 KB: one statevector per wave
  const int lane = threadIdx.x & 31;
  const int wave = threadIdx.x >> 5;
  const int b = blockIdx.x*8 + wave;
  float2* S = st[wave];
  const float* xr = x + (size_t)b*QDIM;

  // amplitude embedding with normalization
  float vals[8]; float ss = 0.f;
  #pragma unroll
  for (int j=0;j<8;++j){ float v = xr[lane + 32*j]; vals[j]=v; ss += v*v; }
  #pragma unroll
  for (int off=16; off; off>>=1) ss += __shfl_xor(ss, off, 32);
  float rn = rsqrtf(ss);
  #pragma unroll
  for (int j=0;j<8;++j) S[lane + 32*j] = make_float2(vals[j]*rn, 0.f);
  // same-wave LDS ops are in-order (DScnt); no cross-wave sharing -> no barriers

  const float2* U = (const float2*)(ws + WS_U);
  for (int l=0; l<NL; ++l){
    #pragma unroll
    for (int q=0; q<NQ; ++q){
      const float2* u = U + (l*NQ + q)*4;
      float2 u00=u[0], u01=u[1], u10=u[2], u11=u[3];
      unsigned s = 1u << (7 - q);           // wire q lives at bit 7-q
      #pragma unroll
      for (int j=0;j<4;++j){                // 128 pairs, 4 per lane
        unsigned p  = lane + 32u*j;
        unsigned i0 = ((p & ~(s-1u)) << 1) | (p & (s-1u));
        unsigned i1 = i0 | s;
        float2 a = S[i0], c = S[i1];
        S[i0] = cadd(cmul(u00,a), cmul(u01,c));
        S[i1] = cadd(cmul(u10,a), cmul(u11,c));
      }
    }
    // layer CNOT permutation: all loads issued before stores; in-order per wave
    float2 tmp[8];
    #pragma unroll
    for (int j=0;j<8;++j) tmp[j] = S[cnot_src(lane + 32u*j)];
    #pragma unroll
    for (int j=0;j<8;++j) S[lane + 32u*j] = tmp[j];
  }

  // Z expectations on wires 0..3 (bits 7..4)
  float z[4] = {0.f,0.f,0.f,0.f};
  #pragma unroll
  for (int j=0;j<8;++j){
    unsigned i = lane + 32u*j;
    float2 a = S[i];
    float p = a.x*a.x + a.y*a.y;
    z[0] += ((i>>7)&1u) ? -p : p;
    z[1] += ((i>>6)&1u) ? -p : p;
    z[2] += ((i>>5)&1u) ? -p : p;
    z[3] += ((i>>4)&1u) ? -p : p;
  }
  #pragma unroll
  for (int w=0; w<4; ++w){
    #pragma unroll
    for (int off=16; off; off>>=1) z[w] += __shfl_xor(z[w], off, 32);
  }
  if (lane == 0){
    float* qz = ws + WS_QZ + (size_t)b*4;
    qz[0]=z[0]; qz[1]=z[1]; qz[2]=z[2]; qz[3]=z[3];
  }
}

// ---------------- K4: adapter + combine + classifier L1 (pre-BN) + stats ----------------
__global__ void __launch_bounds__(256) k_head1(const float* __restrict__ aW,
    const float* __restrict__ ab, const float* __restrict__ cW1,
    const float* __restrict__ cb1, float* __restrict__ ws){
  int b = blockIdx.x*256 + threadIdx.x;
  int lane = threadIdx.x & 31;
  const float* qz = ws + WS_QZ + (size_t)b*4;
  const float* cf = ws + WS_CF + (size_t)b*4;
  float h[4];
  #pragma unroll
  for (int i=0;i<4;++i){
    float s = ab[i];
    #pragma unroll
    for (int j=0;j<4;++j) s += aW[i*4+j]*qz[j];
    h[i] = s + cf[i];
  }
  float* a1 = ws + WS_A1 + (size_t)b*32;
  float* gs = ws + WS_ST1;
  float* gq = ws + WS_ST1 + 32;
  #pragma unroll
  for (int j=0;j<32;++j){
    float s = cb1[j];
    #pragma unroll
    for (int i=0;i<4;++i) s += cW1[j*4+i]*h[i];
    a1[j] = s;
    float q = s*s;
    #pragma unroll
    for (int off=16; off; off>>=1){ s += __shfl_xor(s,off,32); q += __shfl_xor(q,off,32); }
    if (lane==0){ atomicAdd(gs+j, s); atomicAdd(gq+j, q); }
  }
}

// ---------------- K5: BN1 + relu + classifier L2 (pre-BN) + stats ----------------
__global__ void __launch_bounds__(256) k_head2(const float* __restrict__ g1,
    const float* __restrict__ be1, const float* __restrict__ cW2,
    const float* __restrict__ cb2, float* __restrict__ ws){
  int b = blockIdx.x*256 + threadIdx.x;
  int lane = threadIdx.x & 31;
  const float* a1 = ws + WS_A1 + (size_t)b*32;
  const float* gs = ws + WS_ST1;
  const float* gq = ws + WS_ST1 + 32;
  const float invB = 1.f/(float)BATCH;
  float h[32];
  #pragma unroll
  for (int j=0;j<32;++j){
    float m = gs[j]*invB;
    float v = gq[j]*invB - m*m;            // biased variance, matches jnp var
    float xn = g1[j]*(a1[j]-m)*rsqrtf(v+1e-5f) + be1[j];
    h[j] = xn > 0.f ? xn : 0.f;
  }
  float* a2  = ws + WS_A2 + (size_t)b*16;
  float* gs2 = ws + WS_ST2;
  float* gq2 = ws + WS_ST2 + 16;
  #pragma unroll
  for (int j=0;j<16;++j){
    float s = cb2[j];
    #pragma unroll
    for (int i=0;i<32;++i) s += cW2[j*32+i]*h[i];
    a2[j] = s;
    float q = s*s;
    #pragma unroll
    for (int off=16; off; off>>=1){ s += __shfl_xor(s,off,32); q += __shfl_xor(q,off,32); }
    if (lane==0){ atomicAdd(gs2+j, s); atomicAdd(gq2+j, q); }
  }
}

// ---------------- K6: BN2 + relu + L3 relu + L4 sigmoid ----------------
__global__ void __launch_bounds__(256) k_head3(const float* __restrict__ g2,
    const float* __restrict__ be2, const float* __restrict__ cW3,
    const float* __restrict__ cb3, const float* __restrict__ cW4,
    const float* __restrict__ cb4, const float* __restrict__ ws,
    float* __restrict__ out){
  int b = blockIdx.x*256 + threadIdx.x;
  const float* a2 = ws + WS_A2 + (size_t)b*16;
  const float* gs = ws + WS_ST2;
  const float* gq = ws + WS_ST2 + 16;
  const float invB = 1.f/(float)BATCH;
  float h[16];
  #pragma unroll
  for (int j=0;j<16;++j){
    float m = gs[j]*invB;
    float v = gq[j]*invB - m*m;
    float xn = g2[j]*(a2[j]-m)*rsqrtf(v+1e-5f) + be2[j];
    h[j] = xn > 0.f ? xn : 0.f;
  }
  float h3[8];
  #pragma unroll
  for (int j=0;j<8;++j){
    float s = cb3[j];
    #pragma unroll
    for (int i=0;i<16;++i) s += cW3[j*16+i]*h[i];
    h3[j] = s > 0.f ? s : 0.f;
  }
  float o = cb4[0];
  #pragma unroll
  for (int i=0;i<8;++i) o += cW4[i]*h3[i];
  out[b] = 1.f/(1.f + expf(-o));
}

extern "C" void kernel_launch(void* const* d_in, const int* in_sizes, int n_in,
                              void* d_out, int out_size, void* d_ws, size_t ws_size,
                              hipStream_t stream) {
  const float* x   = (const float*)d_in[0];
  const float* qw  = (const float*)d_in[1];
  const float* aW  = (const float*)d_in[2];
  const float* ab  = (const float*)d_in[3];
  const float* fW1 = (const float*)d_in[4];
  const float* fb1 = (const float*)d_in[5];
  const float* fW2 = (const float*)d_in[6];
  const float* fb2 = (const float*)d_in[7];
  const float* cW1 = (const float*)d_in[8];
  const float* cb1 = (const float*)d_in[9];
  const float* g1  = (const float*)d_in[10];
  const float* be1 = (const float*)d_in[11];
  const float* cW2 = (const float*)d_in[12];
  const float* cb2 = (const float*)d_in[13];
  const float* g2  = (const float*)d_in[14];
  const float* be2 = (const float*)d_in[15];
  const float* cW3 = (const float*)d_in[16];
  const float* cb3 = (const float*)d_in[17];
  const float* cW4 = (const float*)d_in[18];
  const float* cb4 = (const float*)d_in[19];
  float* ws  = (float*)d_ws;
  float* out = (float*)d_out;

  // zero the BN-stat accumulators (memset nodes are graph-capturable)
  hipMemsetAsync((char*)d_ws + (size_t)WS_ST1*sizeof(float), 0, 64*sizeof(float), stream);
  hipMemsetAsync((char*)d_ws + (size_t)WS_ST2*sizeof(float), 0, 32*sizeof(float), stream);

  k_fuse_rot<<<1, 64, 0, stream>>>(qw, ws);
  k_feature<<<BATCH/64, 128, 0, stream>>>(x, fW1, fb1, fW2, fb2, ws);
  k_quantum<<<BATCH/8, 256, 0, stream>>>(x, ws);
  k_head1 <<<BATCH/256, 256, 0, stream>>>(aW, ab, cW1, cb1, ws);
  k_head2 <<<BATCH/256, 256, 0, stream>>>(g1, be1, cW2, cb2, ws);
  k_head3 <<<BATCH/256, 256, 0, stream>>>(g2, be2, cW3, cb3, cW4, cb4, ws, out);
}